// MaxMinPointDist_77386720740129
// MI455X (gfx1250) — compile-verified
//
#include <hip/hip_runtime.h>

typedef __attribute__((ext_vector_type(2))) float v2f;
typedef __attribute__((ext_vector_type(8))) float v8f;

#define N_PTS 4096
#define TILES (N_PTS / 16)      // 256 column tiles
#define WPB 8                   // 8 wave32 per 256-thread block

// No NaNs anywhere in this computation: drop IEEE canonicalization so
// fminf/fmaxf lower to bare v_min_num/v_max_num (and fuse into v_min3).
#pragma float_control(push)
#pragma float_control(precise, off)

// Pre-seed d_out[b] with CLAMP=8.0f; the atomic-max then implements max(c, 8).
__global__ void mmpd_init(float* out) {
    if (threadIdx.x < 8) out[threadIdx.x] = 8.0f;
}

__global__ __launch_bounds__(256) void mmpd_kernel(const float* __restrict__ point,
                                                   float* __restrict__ out) {
    __shared__ float4 pts[N_PTS];   // (x, y, z, ||p||^2) : 64 KB LDS

    const int tid = threadIdx.x;
    const int b   = blockIdx.y;
    const float* pb = point + (size_t)b * N_PTS * 3;

    // Stage the whole batch into LDS once, squared norm precomputed.
    for (int i = tid; i < N_PTS; i += 256) {
        float x = pb[3 * i + 0];
        float y = pb[3 * i + 1];
        float z = pb[3 * i + 2];
        pts[i] = make_float4(x, y, z, x * x + y * y + z * z);
    }
    __syncthreads();

    const int  wave = tid >> 5;
    const int  lane = tid & 31;
    const bool lo   = lane < 16;
    const int  l15  = lane & 15;
    const int  row_base = (blockIdx.x * WPB + wave) * 16;
    // Wave-uniform diagonal-tile index -> scalar (SGPR) loop bounds, EXEC
    // stays all-1s around the WMMAs (ISA requirement).
    const int  dt = __builtin_amdgcn_readfirstlane(row_base >> 4);

    // A (16x4 f32): row i = (-2x_i, -2y_i, -2z_i, 1).
    // Lanes 0-15 hold (K0,K1), lanes 16-31 hold (K2,K3).
    float4 qa = pts[row_base + l15];
    v2f a;
    a.x = lo ? (-2.0f * qa.x) : (-2.0f * qa.z);
    a.y = lo ? (-2.0f * qa.y) : 1.0f;

    // B (4x16 f32): col j = (x_j, y_j, z_j, sq_j). Lanes 0-15 need (x,y) = low
    // 8B of pts[j]; lanes 16-31 need (z,sq) = high 8B. One ds_load_b64, no VALU.
    const float2* pts2 = (const float2*)pts;
    const int half = lo ? 0 : 1;

    float minv[8];                              // running min of (sq_j - 2 p_i.p_j)
    #pragma unroll
    for (int t = 0; t < 8; ++t) minv[t] = 1e30f;

    auto tile = [&](int ct) {                   // steady-state tile: no diag, no clamp
        float2 q = pts2[(ct * 16 + l15) * 2 + half];
        v2f bv; bv.x = q.x; bv.y = q.y;
        v8f c = {};                             // SRC2 = inline 0
        v8f d = __builtin_amdgcn_wmma_f32_16x16x4_f32(
            false, a, false, bv, (short)0, c, false, false);
        #pragma unroll
        for (int t = 0; t < 8; ++t) minv[t] = fminf(minv[t], d[t]);
    };

    #pragma unroll 2
    for (int ct = 0; ct < dt; ++ct) tile(ct);

    {   // peeled diagonal tile: mask out the element where col == row
        float2 q = pts2[(dt * 16 + l15) * 2 + half];
        v2f bv; bv.x = q.x; bv.y = q.y;
        v8f c = {};
        v8f d = __builtin_amdgcn_wmma_f32_16x16x4_f32(
            false, a, false, bv, (short)0, c, false, false);
        const int tdiag = l15 - (lo ? 0 : 8);   // in [0,8) only for lanes holding a diag elem
        #pragma unroll
        for (int t = 0; t < 8; ++t) {
            float v = (t == tdiag) ? 1e30f : d[t];
            minv[t] = fminf(minv[t], v);
        }
    }

    #pragma unroll 2
    for (int ct = dt + 1; ct < TILES; ++ct) tile(ct);

    // Row min: the 16 columns of a row live in one 16-lane half.
    #pragma unroll
    for (int m = 1; m <= 8; m <<= 1) {
        #pragma unroll
        for (int t = 0; t < 8; ++t)
            minv[t] = fminf(minv[t], __shfl_xor(minv[t], m, 32));
    }

    // Add back the row-constant sq_i, clamp to 0, max over this wave's rows.
    const int rowv = row_base + (lo ? 0 : 8);
    float vmax2 = 0.0f;
    #pragma unroll
    for (int t = 0; t < 8; ++t) {
        float d2 = fmaxf(minv[t] + pts[rowv + t].w, 0.0f);
        vmax2 = fmaxf(vmax2, d2);
    }
    vmax2 = fmaxf(vmax2, __shfl_xor(vmax2, 16, 32));
    float vmax = sqrtf(vmax2);                  // single sqrt: max∘sqrt = sqrt∘max

    // Distances >= 0 -> IEEE bits monotone as unsigned.
    if (lane == 0)
        atomicMax((unsigned int*)(out + b), __float_as_uint(vmax));
}

#pragma float_control(pop)

extern "C" void kernel_launch(void* const* d_in, const int* in_sizes, int n_in,
                              void* d_out, int out_size, void* d_ws, size_t ws_size,
                              hipStream_t stream) {
    const float* point = (const float*)d_in[0];   // [8, 4096, 3] f32
    float* out = (float*)d_out;                   // [8] f32

    mmpd_init<<<1, 32, 0, stream>>>(out);

    dim3 grid(TILES / WPB, 8);                    // 32 row-tile groups x 8 batches
    mmpd_kernel<<<grid, 256, 0, stream>>>(point, out);
}